// HypergraphLayer_14027363189117
// MI455X (gfx1250) — compile-verified
//
#include <hip/hip_runtime.h>

// CDNA5 / gfx1250, wave32. Three f32 WMMA GEMM stages:
//   theta = X @ W
//   T1    = De^-1 * (H^T @ theta)
//   out   = Dv^-1 * (H  @ T1)
// Diagonals consumed via stride (E+1) reads of the dense D matrices.

typedef float v2f __attribute__((ext_vector_type(2)));
typedef float v8f __attribute__((ext_vector_type(8)));

#define NN 4096
#define EE 4096
#define F_DIM 128

#define UCHUNK 8              // k-steps per pipelined chunk (k advances 32)
#define KCHUNK (4 * UCHUNK)   // 32
#define PF_CHUNKS 4           // prefetch distance in chunks (128 k ahead)

// Load one chunk of A/B fragments (ISA 16x4 / 4x16 f32 layouts; the per-lane
// half-offset is already folded into Ap/Bp, so these are immediate offsets).
template <int ACS>
__device__ __forceinline__ void load_chunk(const float* __restrict__ Ap,
                                           const float* __restrict__ Bp,
                                           v2f (&a)[UCHUNK],
                                           v2f (&b)[UCHUNK]) {
#pragma unroll
  for (int u = 0; u < UCHUNK; ++u) {
    a[u].x = Ap[(long)(4 * u) * ACS];
    a[u].y = Ap[(long)(4 * u + 1) * ACS];
    b[u].x = Bp[(long)(4 * u) * F_DIM];
    b[u].y = Bp[(long)(4 * u + 1) * F_DIM];
  }
}

__device__ __forceinline__ void compute_chunk(const v2f (&a)[UCHUNK],
                                              const v2f (&b)[UCHUNK],
                                              v8f& acc) {
#pragma unroll
  for (int u = 0; u < UCHUNK; ++u)
    acc = __builtin_amdgcn_wmma_f32_16x16x4_f32(
        false, a[u], false, b[u], (short)0, acc, false, false);
}

// C[m, col] = scale(m) * sum_k A(m,k) * B[k, col]
//   A(m,k) = A[m*ARS + k*ACS]  (compile-time strides -> immediate offsets)
//   B is K x 128 row-major; C is M x 128 row-major.
//   SCALE: scale(m) = 1/diag[m*dstride]
// Grid: M/16 blocks of 256 threads (8 waves). Block owns a 16-row tile of C;
// wave j owns 16-column tile j. All 8 waves read identical A fragments ->
// served from WGP$ after the first, so H streams once per GEMM.
// K/KCHUNK must be even (stage K's: 128 -> 4 chunks, 4096 -> 128 chunks).
template <int ARS, int ACS, bool SCALE>
__global__ __launch_bounds__(256) void hg_gemm_f32_wmma(
    const float* __restrict__ A, const float* __restrict__ B,
    float* __restrict__ C, const float* __restrict__ diag, long dstride,
    int K) {
  const int lane = threadIdx.x & 31;
  const int wave = threadIdx.x >> 5;  // 0..7 -> column tile
  const int lm   = lane & 15;         // 0..15
  const int half = lane >> 4;         // 0: K={0,1}  1: K={2,3} per 4-step
  const int m0   = blockIdx.x * 16;
  const int col0 = wave * 16 + lm;

  const float* __restrict__ Ap =
      A + (long)(m0 + lm) * ARS + (long)(2 * half) * ACS;
  const float* __restrict__ Bp = B + col0 + (long)(2 * half) * F_DIM;

  v8f acc = {};
  // Ping-pong fragment buffers: no register rotation copies between chunks.
  v2f a0[UCHUNK], b0[UCHUNK], a1[UCHUNK], b1[UCHUNK];

  load_chunk<ACS>(Ap, Bp, a0, b0);  // chunk 0

  const int chunks = K / KCHUNK;  // even by construction
  for (int c = 1; c < chunks - 1; c += 2) {
    Ap += (long)KCHUNK * ACS;
    Bp += (long)KCHUNK * F_DIM;
    load_chunk<ACS>(Ap, Bp, a1, b1);  // chunk c  (flies over compute below)
    if (c + PF_CHUNKS < chunks)
      __builtin_prefetch(Ap + (long)PF_CHUNKS * KCHUNK * ACS, 0, 3);
    compute_chunk(a0, b0, acc);       // chunk c-1

    Ap += (long)KCHUNK * ACS;
    Bp += (long)KCHUNK * F_DIM;
    load_chunk<ACS>(Ap, Bp, a0, b0);  // chunk c+1
    if (c + 1 + PF_CHUNKS < chunks)
      __builtin_prefetch(Ap + (long)PF_CHUNKS * KCHUNK * ACS, 0, 3);
    compute_chunk(a1, b1, acc);       // chunk c
  }

  // Drain: last chunk loads while chunk (chunks-2) computes.
  Ap += (long)KCHUNK * ACS;
  Bp += (long)KCHUNK * F_DIM;
  load_chunk<ACS>(Ap, Bp, a1, b1);
  compute_chunk(a0, b0, acc);
  compute_chunk(a1, b1, acc);

  // Per-row 1/deg scaling + store. VGPR v holds row m0 + v + 8*half.
#pragma unroll
  for (int v = 0; v < 8; ++v) {
    const int row = m0 + half * 8 + v;
    float s = 1.0f;
    if constexpr (SCALE) s = 1.0f / diag[(long)row * dstride];
    C[(long)row * F_DIM + col0] = acc[v] * s;
  }
}

extern "C" void kernel_launch(void* const* d_in, const int* in_sizes, int n_in,
                              void* d_out, int out_size, void* d_ws,
                              size_t ws_size, hipStream_t stream) {
  (void)in_sizes; (void)n_in; (void)out_size; (void)ws_size;
  const float* X  = (const float*)d_in[0];  // [N, F_IN]
  const float* H  = (const float*)d_in[1];  // [N, E]
  const float* Dv = (const float*)d_in[2];  // [N, N] diagonal
  const float* De = (const float*)d_in[3];  // [E, E] diagonal
  const float* W  = (const float*)d_in[4];  // [F_IN, F_OUT]
  float* out = (float*)d_out;               // [N, F_OUT]

  float* theta = (float*)d_ws;              // [N, F_OUT]  2 MB
  float* T1    = theta + (long)NN * F_DIM;  // [E, F_OUT]  2 MB

  dim3 blk(256);

  // Stage 1: theta = X @ W            (M=N, K=F_IN; A=X row-major)
  hg_gemm_f32_wmma<F_DIM, 1, false><<<NN / 16, blk, 0, stream>>>(
      X, W, theta, nullptr, 0, F_DIM);

  // Stage 2: T1 = De^-1 (H^T @ theta) (M=E, K=N; A(e,n)=H[n*E+e])
  hg_gemm_f32_wmma<1, EE, true><<<EE / 16, blk, 0, stream>>>(
      H, theta, T1, De, (long)EE + 1, NN);

  // Stage 3: out = Dv^-1 (H @ T1)     (M=N, K=E; A=H row-major)
  hg_gemm_f32_wmma<EE, 1, true><<<NN / 16, blk, 0, stream>>>(
      H, T1, out, Dv, (long)NN + 1, EE);
}